// DenseFlashAttention_63410897158644
// MI455X (gfx1250) — compile-verified
//
#include <hip/hip_runtime.h>
#include <math.h>

#define NN 20000
#define NE 200000
#define FD 64
#define NH 4
#define MD 32
#define NTILES (NN / 16) /* 1250 */

typedef __attribute__((ext_vector_type(2))) float v2f;
typedef __attribute__((ext_vector_type(8))) float v8f;

#define WMMA4(ACC, A, B)                                                      \
  (ACC) = __builtin_amdgcn_wmma_f32_16x16x4_f32(false, (A), false, (B),       \
                                                (short)0, (ACC), false, false)

__device__ __forceinline__ bool finitef(float x) {
  return (__float_as_uint(x) & 0x7f800000u) != 0x7f800000u;
}
__device__ __forceinline__ float softplusf(float x) {
  return x > 20.f ? x : log1pf(__expf(x));
}
__device__ __forceinline__ float sigmoidf(float x) {
  return 1.f / (1.f + __expf(-x));
}
__device__ __forceinline__ void atomicMaxF(float* a, float v) {
  unsigned int* ua = (unsigned int*)a;
  unsigned int old = __float_as_uint(*a);
  while (__uint_as_float(old) < v) {
    unsigned int assumed = old;
    old = atomicCAS(ua, assumed, __float_as_uint(v));
    if (old == assumed) break;
  }
}

// ---------------------------------------------------------------------------
// Kernel 0: init scratch (poisoned by harness -> must re-init every launch)
// ---------------------------------------------------------------------------
__global__ __launch_bounds__(256) void init_kernel(float* rmax, float* tmax,
                                                   float* rden, float* tden,
                                                   float* agg) {
  int i = blockIdx.x * 256 + threadIdx.x;
  if (i < NH * NN) {
    rmax[i] = -INFINITY;
    tmax[i] = -INFINITY;
    rden[i] = 0.f;
    tden[i] = 0.f;
  }
  if (i < NN * FD) agg[i] = 0.f;
}

// ---------------------------------------------------------------------------
// Kernel 1: node projections ep/rp/tp[h] = x @ W[h]  (WMMA f32 16x16x4)
// one wave per 16-node row tile per (proj, head); 4 n-tiles * 16 k-steps
// ---------------------------------------------------------------------------
__global__ __launch_bounds__(32) void proj_kernel(
    const float* __restrict__ x, const float* __restrict__ Wp,
    const float* __restrict__ Wr, const float* __restrict__ Wt,
    float* __restrict__ ep, float* __restrict__ rp, float* __restrict__ tp) {
  const int tile = blockIdx.x;
  const int ph = blockIdx.y;  // 0..11
  const int proj = ph >> 2;
  const int h = ph & 3;
  const float* W = (proj == 0 ? Wp : proj == 1 ? Wr : Wt) + h * FD * FD;
  float* out = (proj == 0 ? ep : proj == 1 ? rp : tp) + (size_t)h * NN * FD;

  const int lane = threadIdx.x & 31;
  const int half = lane >> 4;
  const int lm = lane & 15;
  const float* xrow = x + (size_t)(tile * 16 + lm) * FD;

  v8f a0 = {0, 0, 0, 0, 0, 0, 0, 0}, a1 = a0, a2 = a0, a3 = a0;
#pragma unroll
  for (int kt = 0; kt < 16; ++kt) {
    const int k0 = kt * 4 + half * 2;  // lanes 0-15: K=0,1  lanes 16-31: K=2,3
    v2f a;
    a.x = xrow[k0];
    a.y = xrow[k0 + 1];
    const float* w0 = W + k0 * FD;
    v2f b0, b1, b2, b3;
    b0.x = w0[lm];      b0.y = w0[FD + lm];
    b1.x = w0[16 + lm]; b1.y = w0[FD + 16 + lm];
    b2.x = w0[32 + lm]; b2.y = w0[FD + 32 + lm];
    b3.x = w0[48 + lm]; b3.y = w0[FD + 48 + lm];
    WMMA4(a0, a, b0);
    WMMA4(a1, a, b1);
    WMMA4(a2, a, b2);
    WMMA4(a3, a, b3);
  }
  float* ob = out + (size_t)(tile * 16 + half * 8) * FD;
#pragma unroll
  for (int v = 0; v < 8; ++v) {
    ob[v * FD + lm] = a0[v];
    ob[v * FD + 16 + lm] = a1[v];
    ob[v * FD + 32 + lm] = a2[v];
    ob[v * FD + 48 + lm] = a3[v];
  }
}

// ---------------------------------------------------------------------------
// Kernel 2: per-NODE decay/temp offsets (rf = ep[:,receiver] depends only on
// the node!)  off[h][n] = silu(ep[h][n] @ W1[h] + b1[h]) . w2[h] + b2[h]
// WMMA 16x32 tile + in-register silu + cross-lane dot reduction
// ---------------------------------------------------------------------------
__global__ __launch_bounds__(32) void mlp_kernel(
    const float* __restrict__ ep, const float* __restrict__ dW1,
    const float* __restrict__ db1, const float* __restrict__ dw2,
    const float* __restrict__ db2, const float* __restrict__ tW1,
    const float* __restrict__ tb1, const float* __restrict__ tw2,
    const float* __restrict__ tb2, float* __restrict__ doff,
    float* __restrict__ toff) {
  const int tile = blockIdx.x;
  const int h = blockIdx.y;
  const int which = blockIdx.z;  // 0=decay, 1=temp
  const float* W1 = (which ? tW1 : dW1) + h * FD * MD;
  const float* b1 = (which ? tb1 : db1) + h * MD;
  const float* w2 = (which ? tw2 : dw2) + h * MD;
  const float b2 = (which ? tb2 : db2)[h];
  float* off = (which ? toff : doff) + h * NN;

  const int lane = threadIdx.x & 31;
  const int half = lane >> 4;
  const int lm = lane & 15;
  const float* arow = ep + ((size_t)h * NN + tile * 16 + lm) * FD;

  v8f c0 = {0, 0, 0, 0, 0, 0, 0, 0}, c1 = c0;
#pragma unroll
  for (int kt = 0; kt < 16; ++kt) {
    const int k0 = kt * 4 + half * 2;
    v2f a;
    a.x = arow[k0];
    a.y = arow[k0 + 1];
    const float* w0 = W1 + k0 * MD;
    v2f b0, b1v;
    b0.x = w0[lm];       b0.y = w0[MD + lm];
    b1v.x = w0[16 + lm]; b1v.y = w0[MD + 16 + lm];
    WMMA4(c0, a, b0);
    WMMA4(c1, a, b1v);
  }
  const float bias0 = b1[lm], bias1 = b1[16 + lm];
  const float g0 = w2[lm], g1 = w2[16 + lm];
  v8f part;
#pragma unroll
  for (int v = 0; v < 8; ++v) {
    float s0 = c0[v] + bias0;
    s0 = s0 * sigmoidf(s0);  // silu
    float s1 = c1[v] + bias1;
    s1 = s1 * sigmoidf(s1);
    part[v] = s0 * g0 + s1 * g1;
  }
  // reduce the 16 lanes of each half-wave (rows live per-VGPR)
#pragma unroll
  for (int m = 1; m < 16; m <<= 1)
#pragma unroll
    for (int v = 0; v < 8; ++v) part[v] += __shfl_xor(part[v], m, 32);
  if (lm == 0) {
    const int rbase = tile * 16 + half * 8;
#pragma unroll
    for (int v = 0; v < 8; ++v) off[rbase + v] = part[v] + b2;
  }
}

// ---------------------------------------------------------------------------
// Kernel 3: per-edge logits (one wave per edge, lanes split the 64 features)
// + segmented atomic max over receivers
// ---------------------------------------------------------------------------
__global__ __launch_bounds__(256) void logits_kernel(
    const int* __restrict__ ei, const float* __restrict__ elen,
    const float* __restrict__ ep, const float* __restrict__ rs,
    const float* __restrict__ ts, const float* __restrict__ rdls,
    const float* __restrict__ rtb, const float* __restrict__ rtw,
    const float* __restrict__ doff, const float* __restrict__ toff,
    float* __restrict__ rlog, float* __restrict__ tlog,
    float* __restrict__ rmax, float* __restrict__ tmax) {
  const int e = blockIdx.x * 8 + (threadIdx.x >> 5);
  if (e >= NE) return;
  const int lane = threadIdx.x & 31;
  const int s = ei[e];
  const int r = ei[NE + e];
  const float len = elen[e];
  const int f0 = lane * 2;
#pragma unroll
  for (int h = 0; h < NH; ++h) {
    const float* epb = ep + (size_t)h * NN * FD;
    const float es0 = epb[(size_t)s * FD + f0], es1 = epb[(size_t)s * FD + f0 + 1];
    const float er0 = epb[(size_t)r * FD + f0], er1 = epb[(size_t)r * FD + f0 + 1];
    const float d0 = es0 - er0, d1 = es1 - er1;
    float dr = d0 * rs[h * FD + f0] + d1 * rs[h * FD + f0 + 1];
    float dt = d0 * ts[h * FD + f0] + d1 * ts[h * FD + f0 + 1];
#pragma unroll
    for (int m = 1; m < 32; m <<= 1) {
      dr += __shfl_xor(dr, m, 32);
      dt += __shfl_xor(dt, m, 32);
    }
    if (lane == 0) {
      const float rdscale = softplusf(rdls[h]);
      float rl = dr - (rdscale + doff[h * NN + r]) * len;
      const float rtemp = softplusf(rtb[h] + rtw[h] * len + toff[h * NN + r]);
      rl = rl / (rtemp + 1e-4f);
      rlog[(size_t)h * NE + e] = rl;
      tlog[(size_t)h * NE + e] = dt;
      atomicMaxF(&rmax[h * NN + r], rl);
      atomicMaxF(&tmax[h * NN + r], dt);
    }
  }
}

// ---------------------------------------------------------------------------
// Kernel 4: e = exp(l - max), segmented sum of denominators
// ---------------------------------------------------------------------------
__global__ __launch_bounds__(256) void expden_kernel(
    const int* __restrict__ ei, const float* __restrict__ rlog,
    const float* __restrict__ tlog, const float* __restrict__ rmax,
    const float* __restrict__ tmax, float* __restrict__ rexp,
    float* __restrict__ texp, float* __restrict__ rden,
    float* __restrict__ tden) {
  const int idx = blockIdx.x * 256 + threadIdx.x;
  if (idx >= NH * NE) return;
  const int h = idx / NE;
  const int e = idx - h * NE;
  const int r = ei[NE + e];
  float mr = rmax[h * NN + r];
  if (!finitef(mr)) mr = 0.f;
  float mt = tmax[h * NN + r];
  if (!finitef(mt)) mt = 0.f;
  const float er = __expf(rlog[idx] - mr);
  const float et = __expf(tlog[idx] - mt);
  rexp[idx] = er;
  texp[idx] = et;
  atomicAdd(&rden[h * NN + r], er);
  atomicAdd(&tden[h * NN + r], et);
}

// ---------------------------------------------------------------------------
// Kernel 5: gated aggregation: one wave per edge; sum over heads in registers
// then 2 float atomics per lane into agg[node][f]
// ---------------------------------------------------------------------------
__global__ __launch_bounds__(256) void agg_kernel(
    const int* __restrict__ ei, const float* __restrict__ elen,
    const float* __restrict__ rp, const float* __restrict__ tp,
    const float* __restrict__ rexp, const float* __restrict__ texp,
    const float* __restrict__ rden, const float* __restrict__ tden,
    const float* __restrict__ mixb, const float* __restrict__ mixs,
    float* __restrict__ agg) {
  const int e = blockIdx.x * 8 + (threadIdx.x >> 5);
  if (e >= NE) return;
  const int lane = threadIdx.x & 31;
  const int s = ei[e];
  const int r = ei[NE + e];
  const float len = elen[e];
  const int f0 = lane * 2;
  float acc0 = 0.f, acc1 = 0.f;
#pragma unroll
  for (int h = 0; h < NH; ++h) {
    const float* rpb = rp + (size_t)h * NN * FD;
    const float* tpb = tp + (size_t)h * NN * FD;
    __builtin_prefetch(tpb + (size_t)s * FD + f0, 0, 0);
    __builtin_prefetch(tpb + (size_t)r * FD + f0, 0, 0);
    float ra = rexp[(size_t)h * NE + e] / (rden[h * NN + r] + 1e-9f);
    if (!finitef(ra)) ra = 0.f;
    float ta = texp[(size_t)h * NE + e] / (tden[h * NN + r] + 1e-9f);
    if (!finitef(ta)) ta = 0.f;
    const float g = sigmoidf(mixb[h] + mixs[h] * len);
    const float ba = g * ra + (1.f - g) * ta;
    const float rd0 = rpb[(size_t)s * FD + f0] - rpb[(size_t)r * FD + f0];
    const float rd1 = rpb[(size_t)s * FD + f0 + 1] - rpb[(size_t)r * FD + f0 + 1];
    const float td0 = tpb[(size_t)s * FD + f0] - tpb[(size_t)r * FD + f0];
    const float td1 = tpb[(size_t)s * FD + f0 + 1] - tpb[(size_t)r * FD + f0 + 1];
    acc0 += ba * (g * rd0 + (1.f - g) * td0);
    acc1 += ba * (g * rd1 + (1.f - g) * td1);
  }
  if (!finitef(acc0)) acc0 = 0.f;
  if (!finitef(acc1)) acc1 = 0.f;
  atomicAdd(&agg[(size_t)r * FD + f0], acc0 * (1.0f / NH));
  atomicAdd(&agg[(size_t)r * FD + f0 + 1], acc1 * (1.0f / NH));
}

// ---------------------------------------------------------------------------
// Kernel 6: out = x + agg @ Wout  (WMMA f32 16x16x4, fused residual)
// ---------------------------------------------------------------------------
__global__ __launch_bounds__(32) void out_kernel(const float* __restrict__ agg,
                                                 const float* __restrict__ x,
                                                 const float* __restrict__ Wout,
                                                 float* __restrict__ out) {
  const int tile = blockIdx.x;
  const int lane = threadIdx.x & 31;
  const int half = lane >> 4;
  const int lm = lane & 15;
  const float* arow = agg + (size_t)(tile * 16 + lm) * FD;

  v8f a0 = {0, 0, 0, 0, 0, 0, 0, 0}, a1 = a0, a2 = a0, a3 = a0;
#pragma unroll
  for (int kt = 0; kt < 16; ++kt) {
    const int k0 = kt * 4 + half * 2;
    v2f a;
    a.x = arow[k0];
    a.y = arow[k0 + 1];
    const float* w0 = Wout + k0 * FD;
    v2f b0, b1, b2, b3;
    b0.x = w0[lm];      b0.y = w0[FD + lm];
    b1.x = w0[16 + lm]; b1.y = w0[FD + 16 + lm];
    b2.x = w0[32 + lm]; b2.y = w0[FD + 32 + lm];
    b3.x = w0[48 + lm]; b3.y = w0[FD + 48 + lm];
    WMMA4(a0, a, b0);
    WMMA4(a1, a, b1);
    WMMA4(a2, a, b2);
    WMMA4(a3, a, b3);
  }
  const size_t rbase = (size_t)(tile * 16 + half * 8) * FD;
#pragma unroll
  for (int v = 0; v < 8; ++v) {
    out[rbase + v * FD + lm]      = x[rbase + v * FD + lm] + a0[v];
    out[rbase + v * FD + 16 + lm] = x[rbase + v * FD + 16 + lm] + a1[v];
    out[rbase + v * FD + 32 + lm] = x[rbase + v * FD + 32 + lm] + a2[v];
    out[rbase + v * FD + 48 + lm] = x[rbase + v * FD + 48 + lm] + a3[v];
  }
}

// ---------------------------------------------------------------------------
extern "C" void kernel_launch(void* const* d_in, const int* in_sizes, int n_in,
                              void* d_out, int out_size, void* d_ws,
                              size_t ws_size, hipStream_t stream) {
  const float* x = (const float*)d_in[0];
  const int* ei = (const int*)d_in[1];
  // d_in[2] = edge_vec (unused by the reference math)
  const float* elen = (const float*)d_in[3];
  const float* Wp = (const float*)d_in[4];
  const float* Wr = (const float*)d_in[5];
  const float* Wt = (const float*)d_in[6];
  const float* rs = (const float*)d_in[7];
  const float* ts = (const float*)d_in[8];
  const float* rdls = (const float*)d_in[9];
  const float* rtb = (const float*)d_in[10];
  const float* rtw = (const float*)d_in[11];
  const float* mixb = (const float*)d_in[12];
  const float* mixs = (const float*)d_in[13];
  const float* dW1 = (const float*)d_in[14];
  const float* db1 = (const float*)d_in[15];
  const float* dw2 = (const float*)d_in[16];
  const float* db2 = (const float*)d_in[17];
  const float* tW1 = (const float*)d_in[18];
  const float* tb1 = (const float*)d_in[19];
  const float* tw2 = (const float*)d_in[20];
  const float* tb2 = (const float*)d_in[21];
  const float* Wout = (const float*)d_in[22];

  float* ws = (float*)d_ws;
  size_t o = 0;
  float* ep = ws + o;   o += (size_t)NH * NN * FD;
  float* rp = ws + o;   o += (size_t)NH * NN * FD;
  float* tp = ws + o;   o += (size_t)NH * NN * FD;
  float* doff = ws + o; o += (size_t)NH * NN;
  float* toff = ws + o; o += (size_t)NH * NN;
  float* rlog = ws + o; o += (size_t)NH * NE;
  float* tlog = ws + o; o += (size_t)NH * NE;
  float* rexp = ws + o; o += (size_t)NH * NE;
  float* texp = ws + o; o += (size_t)NH * NE;
  float* rmax = ws + o; o += (size_t)NH * NN;
  float* tmax = ws + o; o += (size_t)NH * NN;
  float* rden = ws + o; o += (size_t)NH * NN;
  float* tden = ws + o; o += (size_t)NH * NN;
  float* agg = ws + o;  o += (size_t)NN * FD;

  init_kernel<<<(NN * FD + 255) / 256, 256, 0, stream>>>(rmax, tmax, rden,
                                                         tden, agg);
  proj_kernel<<<dim3(NTILES, 12), 32, 0, stream>>>(x, Wp, Wr, Wt, ep, rp, tp);
  mlp_kernel<<<dim3(NTILES, NH, 2), 32, 0, stream>>>(ep, dW1, db1, dw2, db2,
                                                     tW1, tb1, tw2, tb2, doff,
                                                     toff);
  logits_kernel<<<NE / 8, 256, 0, stream>>>(ei, elen, ep, rs, ts, rdls, rtb,
                                            rtw, doff, toff, rlog, tlog, rmax,
                                            tmax);
  expden_kernel<<<(NH * NE + 255) / 256, 256, 0, stream>>>(
      ei, rlog, tlog, rmax, tmax, rexp, texp, rden, tden);
  agg_kernel<<<NE / 8, 256, 0, stream>>>(ei, elen, rp, tp, rexp, texp, rden,
                                         tden, mixb, mixs, agg);
  out_kernel<<<NTILES, 32, 0, stream>>>(agg, x, Wout, (float*)d_out);
}